// dtcwtLayer_81956565942381
// MI455X (gfx1250) — compile-verified
//
#include <hip/hip_runtime.h>

// ---------------------------------------------------------------------------
// DTCWT forward (2 levels) for x = (16, 512, 512, 3) fp32 on MI455X (gfx1250).
// Level-1 separable filters run as banded-matrix matmuls on
// V_WMMA_F32_16X16X4_F32 (fp32-exact), with q2c band packing fused in-register
// via lane shuffles. Level-2 coldfilt computes 2x2 quads per thread with q2c
// fused. ~550MB total traffic vs 250MB roofline (~11us) at 23.3 TB/s.
// ---------------------------------------------------------------------------

typedef float v2f __attribute__((ext_vector_type(2)));
typedef float v8f __attribute__((ext_vector_type(8)));

#define HH   512
#define WCH  512
#define UU   1536               /* W*C                 */
#define IMG  786432             /* H*U (one batch img) */

static constexpr float kH0B[10] = {
  0.03516384f, 0.0f, -0.08832942f, 0.23389032f, 0.76027237f,
  0.5875183f,  0.0f, -0.11430184f, 0.0f,        0.0f};

// symmetric reflection into [0, n)  (single reflection suffices: |overrun|<=20)
__device__ __forceinline__ int refl(int i, int n) {
  i = (i < 0) ? (-1 - i) : i;
  i = (i >= n) ? (2 * n - 1 - i) : i;
  return i;
}

// H0O coefficient for (input - output) offset d (5-tap, symmetric)
__device__ __forceinline__ float coefLo(int d) {
  float r = 0.0f;
  r = (d == -2) ? -0.05f : r;
  r = (d == -1) ?  0.25f : r;
  r = (d ==  0) ?  0.50f : r;
  r = (d ==  1) ?  0.25f : r;
  r = (d ==  2) ? -0.05f : r;
  return r;
}

// H1O coefficient (7-tap, symmetric), /280
__device__ __forceinline__ float coefHi(int d) {
  const float a3 =   -3.0f / 280.0f;
  const float a2 =   15.0f / 280.0f;
  const float a1 =   73.0f / 280.0f;
  const float a0 = -170.0f / 280.0f;
  float r = 0.0f;
  r = (d == -3 || d == 3) ? a3 : r;
  r = (d == -2 || d == 2) ? a2 : r;
  r = (d == -1 || d == 1) ? a1 : r;
  r = (d ==  0)           ? a0 : r;
  return r;
}

__device__ __forceinline__ v8f wmma4(v2f a, v2f b, v8f c) {
  // (neg_a, A, neg_b, B, c_mod, C, reuse_a, reuse_b)
  return __builtin_amdgcn_wmma_f32_16x16x4_f32(false, a, false, b, (short)0, c,
                                               false, false);
}

// ---------------------------------------------------------------------------
// Level-1 vertical filter: lo = H0O*x, hi = H1O*x along H.
// D[m][n] = sum_k A[m][k]*B[k][n]; A(16x4)=const filter chunk, B(4x16)=data
// (coalesced 64B per half-wave). Each wave does 8 row-tiles with rolling
// chunk reuse (4 new chunk loads per tile instead of 6).
// ---------------------------------------------------------------------------
__global__ __launch_bounds__(32) void k_vfilt(const float* __restrict__ x,
                                              float* __restrict__ lo,
                                              float* __restrict__ hi) {
  const int lane = threadIdx.x;
  const int n    = lane & 15;
  const int half = lane >> 4;           // 0: K={0,1}  1: K={2,3}
  const int k0   = half * 2;
  const int uc   = blockIdx.x;          // 0..95
  const int hg   = blockIdx.y;          // 0..3 (8 tiles each)
  const int b    = blockIdx.z;          // 0..15

  v2f aLo[6], aHi[6];
#pragma unroll
  for (int c = 0; c < 6; ++c) {
    const int d0 = 4 * c + k0 - 4 - n;  // (input row offset) - (output row m=n)
    v2f al, ah;
    al.x = coefLo(d0); al.y = coefLo(d0 + 1);
    ah.x = coefHi(d0); ah.y = coefHi(d0 + 1);
    aLo[c] = al; aHi[c] = ah;
  }

  const float* xb  = x  + (size_t)b * IMG + uc * 16 + n;
  float*       lop = lo + (size_t)b * IMG + uc * 16 + n;
  float*       hip = hi + (size_t)b * IMG + uc * 16 + n;

  v2f bd[6];
#pragma unroll
  for (int c = 0; c < 6; ++c) {
    const int r0 = hg * 128 - 4 + 4 * c + k0;
    bd[c].x = xb[(size_t)refl(r0,     HH) * UU];
    bd[c].y = xb[(size_t)refl(r0 + 1, HH) * UU];
  }

  for (int it = 0; it < 8; ++it) {
    const int hbase = hg * 128 + it * 16;
    v8f accLo = {};
    v8f accHi = {};
#pragma unroll
    for (int c = 0; c < 6; ++c) {
      accLo = wmma4(aLo[c], bd[c], accLo);
      accHi = wmma4(aHi[c], bd[c], accHi);
    }
#pragma unroll
    for (int v = 0; v < 8; ++v) {
      const size_t off = (size_t)(hbase + v + half * 8) * UU;
      lop[off] = accLo[v];
      hip[off] = accHi[v];
    }
    if (it < 7) {               // roll: chunks 4,5 -> 0,1 ; load 4 new chunks
      bd[0] = bd[4];
      bd[1] = bd[5];
#pragma unroll
      for (int c = 2; c < 6; ++c) {
        const int r0 = hbase + 16 - 4 + 4 * c + k0;
        bd[c].x = xb[(size_t)refl(r0,     HH) * UU];
        bd[c].y = xb[(size_t)refl(r0 + 1, HH) * UU];
      }
    }
  }
}

// ---------------------------------------------------------------------------
// Fused q2c store for level 1: acc holds a 16(h) x 16(w) tile of one channel.
// h pairs = acc[v],acc[v+1] (same lane); w pairs = adjacent lanes (shfl_xor 1).
// out_h0[b][hh][ww][(ri*3 + c)*6 + band], last dim 36.
// ---------------------------------------------------------------------------
__device__ __forceinline__ void q2c_store_h0(v8f acc, int n, int half, int hbase,
                                             int wbase, int ch, int b, int bp,
                                             int bq, float* __restrict__ oh0) {
  const float s = 0.7071067811865476f;
#pragma unroll
  for (int v = 0; v < 8; v += 2) {
    const float e0 = acc[v];
    const float e1 = acc[v + 1];
    const float o0 = __shfl_xor(e0, 1, 32);   // partner lane: w+1
    const float o1 = __shfl_xor(e1, 1, 32);
    if ((n & 1) == 0) {
      const int hh = (hbase + v + half * 8) >> 1;
      const int ww = (wbase + n) >> 1;
      float* o = oh0 + ((size_t)((b * 256 + hh) * 256 + ww)) * 36 + ch * 6;
      o[bp]      = (e0 - o1) * s;   // real band p: a - d
      o[18 + bp] = (o0 + e1) * s;   // imag band p: b + c
      o[bq]      = (e0 + o1) * s;   // real band q: a + d
      o[18 + bq] = (o0 - e1) * s;   // imag band q: b - c
    }
  }
}

// ---------------------------------------------------------------------------
// Level-1 horizontal filter, both filters per pass from shared A-data.
//   D[m=h][n=w_out]: A(16x4)=data (per-channel, taps along w), B(4x16)=const.
// ISHI=1 (input hi):  H0O -> q2c bands (0,5); H1O -> q2c bands (1,4).
// ISHI=0 (input lo):  H0O -> lolo buffer;     H1O -> q2c bands (2,3).
// Each wave: 8 w-tiles with rolling chunk reuse.
// ---------------------------------------------------------------------------
template <int ISHI>
__global__ __launch_bounds__(32) void k_hpair(const float* __restrict__ in,
                                              float* __restrict__ lolo,
                                              float* __restrict__ oh0) {
  const int lane = threadIdx.x;
  const int n    = lane & 15;
  const int half = lane >> 4;
  const int k0   = half * 2;
  const int wg   = blockIdx.x / 3;      // 0..3 (8 tiles each)
  const int ch   = blockIdx.x % 3;      // 0..2
  const int hb   = blockIdx.y;          // 0..31
  const int b    = blockIdx.z;          // 0..15
  const int hbase = hb * 16;

  v2f bcLo[6], bcHi[6];
#pragma unroll
  for (int c = 0; c < 6; ++c) {
    const int d0 = 4 * c + k0 - 4 - n;  // (input w offset) - (output w = n)
    v2f bl, bh;
    bl.x = coefLo(d0); bl.y = coefLo(d0 + 1);
    bh.x = coefHi(d0); bh.y = coefHi(d0 + 1);
    bcLo[c] = bl; bcHi[c] = bh;
  }

  const float* ip = in + (size_t)b * IMG + (size_t)(hbase + n) * UU + ch;

  v2f ad[6];
#pragma unroll
  for (int c = 0; c < 6; ++c) {
    const int vw0 = wg * 128 - 4 + 4 * c + k0;
    ad[c].x = ip[(size_t)refl(vw0,     WCH) * 3];
    ad[c].y = ip[(size_t)refl(vw0 + 1, WCH) * 3];
  }

  for (int it = 0; it < 8; ++it) {
    const int wbase = wg * 128 + it * 16;
    v8f accLo = {};
    v8f accHi = {};
#pragma unroll
    for (int c = 0; c < 6; ++c) {
      accLo = wmma4(ad[c], bcLo[c], accLo);
      accHi = wmma4(ad[c], bcHi[c], accHi);
    }
    if (ISHI) {
      q2c_store_h0(accLo, n, half, hbase, wbase, ch, b, 0, 5, oh0);  // hi_lo
      q2c_store_h0(accHi, n, half, hbase, wbase, ch, b, 1, 4, oh0);  // hi_hi
    } else {
      float* op = lolo + (size_t)b * IMG + (size_t)(wbase + n) * 3 + ch;
#pragma unroll
      for (int v = 0; v < 8; ++v)
        op[(size_t)(hbase + v + half * 8) * UU] = accLo[v];          // lolo
      q2c_store_h0(accHi, n, half, hbase, wbase, ch, b, 2, 3, oh0);  // lo_hi
    }
    if (it < 7) {
      ad[0] = ad[4];
      ad[1] = ad[5];
#pragma unroll
      for (int c = 2; c < 6; ++c) {
        const int vw0 = wbase + 16 - 4 + 4 * c + k0;
        ad[c].x = ip[(size_t)refl(vw0,     WCH) * 3];
        ad[c].y = ip[(size_t)refl(vw0 + 1, WCH) * 3];
      }
    }
  }
}

// ---------------------------------------------------------------------------
// coldfilt along H: lolo(16,512,512,3) -> lo2,hi2 (16,256,512,3).
//  lo2[2q]  = sum H0B[j]*x[ref(4q+10-2j)]     lo2[2q+1] = sum H0A[j]*x[ref(4q+11-2j)]
//  hi2[2q]  = sum s_j*H0B[j]*x[ref(4q+11-2j)] hi2[2q+1] = sum -s_j*H0A[j]*x[ref(4q+10-2j)]
//  (s_j = +1 even j, -1 odd j; H0A[j] = H0B[9-j])
// ---------------------------------------------------------------------------
__global__ void k_cdf_v(const float* __restrict__ in, float* __restrict__ oLo,
                        float* __restrict__ oHi) {
  const int t = blockIdx.x * blockDim.x + threadIdx.x;
  if (t >= 16 * 256 * 1536) return;
  const int u = t % UU;
  const int i = (t / UU) % 256;
  const int b = t / (256 * UU);
  const int q = i >> 1;
  const int p = i & 1;

  const float* ib = in + (size_t)b * IMG + u;
  float xa[10], xb[10];
#pragma unroll
  for (int j = 0; j < 10; ++j) {
    const int ra = 4 * q + 10 - 2 * j;
    xa[j] = ib[(size_t)refl(ra,     HH) * UU];
    xb[j] = ib[(size_t)refl(ra + 1, HH) * UU];
  }
  float lo = 0.f, hi = 0.f;
#pragma unroll
  for (int j = 0; j < 10; ++j) {
    const float hb = kH0B[j];
    const float ha = kH0B[9 - j];                        // H0A[j]
    const float cl = p ? ha : hb;
    const float chh = p ? ((j & 1) ? ha : -ha) : ((j & 1) ? -hb : hb);
    const float vl = p ? xb[j] : xa[j];
    const float vh = p ? xa[j] : xb[j];
    lo += cl * vl;
    hi += chh * vh;
  }
  const size_t oidx = (size_t)b * 393216 + (size_t)i * UU + u;
  oLo[oidx] = lo;
  oHi[oidx] = hi;
}

// four coldfilt dot products (both variants, both output parities) for one row
__device__ __forceinline__ void cdf_dots(const float xa[10], const float xb[10],
                                         float& vH0e, float& vH0o, float& vH1e,
                                         float& vH1o) {
  vH0e = vH0o = vH1e = vH1o = 0.f;
#pragma unroll
  for (int j = 0; j < 10; ++j) {
    const float hb = kH0B[j];
    const float ha = kH0B[9 - j];
    vH0e += hb * xa[j];
    vH0o += ha * xb[j];
    vH1e += ((j & 1) ? -hb : hb) * xb[j];
    vH1o += ((j & 1) ? ha : -ha) * xa[j];
  }
}

// ---------------------------------------------------------------------------
// coldfilt along W + fused q2c, one 2x2 output quad per thread.
// in (16,256,512,3). LO: write lolo2 (H0 variant) + q2c H1 variant bands(2,3).
//                    HI: q2c H0 variant bands (0,5), H1 variant bands (1,4).
// out_h1[b][hh][ww][(ri*6 + band)*3 + c], last dim 36.
// ---------------------------------------------------------------------------
template <int ISHI>
__global__ void k_cdf2(const float* __restrict__ in, float* __restrict__ lolo2,
                       float* __restrict__ oh1) {
  const int t = blockIdx.x * blockDim.x + threadIdx.x;
  if (t >= 16 * 128 * 128 * 3) return;
  const int c  = t % 3;
  const int ww = (t / 3) % 128;
  const int hh = (t / 384) % 128;
  const int b  = t / 49152;

  const float* r0p = in + (size_t)b * 393216 + (size_t)(2 * hh) * UU + c;
  const float* r1p = r0p + UU;
  float xa0[10], xb0[10], xa1[10], xb1[10];
#pragma unroll
  for (int j = 0; j < 10; ++j) {
    const int wa = 4 * ww + 10 - 2 * j;
    const int ia  = refl(wa,     WCH) * 3;
    const int ib2 = refl(wa + 1, WCH) * 3;
    xa0[j] = r0p[ia];  xb0[j] = r0p[ib2];
    xa1[j] = r1p[ia];  xb1[j] = r1p[ib2];
  }
  float a0e, a0o, b0e, b0o;   // row 2hh   : H0 even/odd, H1 even/odd
  float a1e, a1o, b1e, b1o;   // row 2hh+1
  cdf_dots(xa0, xb0, a0e, a0o, b0e, b0o);
  cdf_dots(xa1, xb1, a1e, a1o, b1e, b1o);

  const float s = 0.7071067811865476f;
  float* o = oh1 + ((size_t)((b * 128 + hh) * 128 + ww)) * 36 + c;

  if (ISHI) {
    // hi_lo = H0 variant -> bands (0,5)
    o[(0 * 6 + 0) * 3] = (a0e - a1o) * s;
    o[(1 * 6 + 0) * 3] = (a0o + a1e) * s;
    o[(0 * 6 + 5) * 3] = (a0e + a1o) * s;
    o[(1 * 6 + 5) * 3] = (a0o - a1e) * s;
    // hi_hi = H1 variant -> bands (1,4)
    o[(0 * 6 + 1) * 3] = (b0e - b1o) * s;
    o[(1 * 6 + 1) * 3] = (b0o + b1e) * s;
    o[(0 * 6 + 4) * 3] = (b0e + b1o) * s;
    o[(1 * 6 + 4) * 3] = (b0o - b1e) * s;
  } else {
    // lolo2 direct store (H0 variant quad)
    float* d0 = lolo2 + (size_t)b * 196608 + (size_t)(2 * hh) * 768 +
                (size_t)(2 * ww) * 3 + c;
    d0[0]       = a0e;
    d0[3]       = a0o;
    d0[768]     = a1e;
    d0[768 + 3] = a1o;
    // lo_hi = H1 variant -> bands (2,3)
    o[(0 * 6 + 2) * 3] = (b0e - b1o) * s;
    o[(1 * 6 + 2) * 3] = (b0o + b1e) * s;
    o[(0 * 6 + 3) * 3] = (b0e + b1o) * s;
    o[(1 * 6 + 3) * 3] = (b0o - b1e) * s;
  }
}

// ---------------------------------------------------------------------------
extern "C" void kernel_launch(void* const* d_in, const int* in_sizes, int n_in,
                              void* d_out, int out_size, void* d_ws, size_t ws_size,
                              hipStream_t stream) {
  const float* x = (const float*)d_in[0];
  float* out = (float*)d_out;
  float* ws  = (float*)d_ws;

  // workspace (floats); peak = 37,748,736 floats = 151 MB
  float* lo   = ws;                  // 12,582,912
  float* hi   = ws + 12582912;       // 12,582,912
  float* lolo = ws + 25165824;       // 12,582,912
  float* lo2  = lo;                  // reuse after level 1 (6,291,456)
  float* hi2  = hi;

  float* lolo2  = out;               //  3,145,728
  float* out_h0 = out + 3145728;     // 37,748,736
  float* out_h1 = out + 40894464;    //  9,437,184

  const dim3 gV(96, 4, 16), gH(12, 32, 16), b32(32), b256(256);

  // level 1 (WMMA): vertical, then horizontal pairs with fused q2c
  k_vfilt<<<gV, b32, 0, stream>>>(x, lo, hi);
  k_hpair<1><<<gH, b32, 0, stream>>>(hi, nullptr, out_h0);   // hi_lo + hi_hi
  k_hpair<0><<<gH, b32, 0, stream>>>(lo, lolo, out_h0);      // lolo + lo_hi

  // level 2: coldfilt along H, then coldfilt along W with fused q2c
  k_cdf_v<<<24576, b256, 0, stream>>>(lolo, lo2, hi2);
  k_cdf2<0><<<3072, b256, 0, stream>>>(lo2, lolo2, out_h1);  // lolo2 + lo_hi
  k_cdf2<1><<<3072, b256, 0, stream>>>(hi2, nullptr, out_h1);// hi_lo + hi_hi
}